// TDGAT_66881230733644
// MI455X (gfx1250) — compile-verified
//
#include <hip/hip_runtime.h>
#include <math.h>

// Problem sizes (fixed by the reference)
#define DD 64
static const int cNu = 100000, cNi = 100000, cNb = 5000, cNc = 2000, cNr = 50000;
static const int cE = 600000, cEb = 100000, cEc = 200000, cEr = 400000;

typedef float v2f __attribute__((ext_vector_type(2)));
typedef float v8f __attribute__((ext_vector_type(8)));

// ---- monotone float <-> uint key for atomicMax on floats ----
__device__ __forceinline__ unsigned fkey(float f) {
  unsigned u = __float_as_uint(f);
  return (u & 0x80000000u) ? ~u : (u | 0x80000000u);
}
__device__ __forceinline__ float funkey(unsigned k) {
  return __uint_as_float((k & 0x80000000u) ? (k & 0x7fffffffu) : ~k);
}
#define NEGINF_KEY 0x007FFFFFu  // fkey(-inf)

__device__ __forceinline__ float wave_sum32(float v) {
#pragma unroll
  for (int off = 16; off > 0; off >>= 1) v += __shfl_xor(v, off, 32);
  return v;
}

// ------------------------- fills -------------------------
__global__ __launch_bounds__(256) void k_fill_f32(float* p, int n, float v) {
  int i = blockIdx.x * 256 + threadIdx.x;
  if (i < n) p[i] = v;
}
__global__ __launch_bounds__(256) void k_fill_u32(unsigned* p, int n, unsigned v) {
  int i = blockIdx.x * 256 + threadIdx.x;
  if (i < n) p[i] = v;
}

// ------------------------- WMMA GEMM: Y[N,64] = X[N,64] @ W[64,64] (+ Add) ----
// One wave -> 16 rows x 64 cols. 4 f32 accumulators, K unrolled in steps of 4.
__global__ __launch_bounds__(256) void k_gemm64(const float* __restrict__ X,
                                                const float* __restrict__ W,
                                                const float* __restrict__ Add,
                                                float* __restrict__ Y, int N) {
  const int lane = threadIdx.x & 31;
  const int wave = threadIdx.x >> 5;  // 0..7
  const int hf = lane >> 4;           // lane half (0/1)
  const int lr = lane & 15;
  const int m0 = (blockIdx.x * 8 + wave) * 16;
  if (m0 >= N) return;  // uniform per wave: EXEC stays all-ones for WMMA

  int arow = m0 + lr;
  if (arow > N - 1) arow = N - 1;  // clamp tail rows (stores are guarded)
  const float* xr = X + (size_t)arow * DD;

  v8f acc0 = {0.f, 0.f, 0.f, 0.f, 0.f, 0.f, 0.f, 0.f};
  v8f acc[4];
  acc[0] = acc0; acc[1] = acc0; acc[2] = acc0; acc[3] = acc0;

#pragma unroll
  for (int kb = 0; kb < DD; kb += 4) {
    // A fragment (16x4 f32): VGPR0=K(2*hf), VGPR1=K(2*hf+1)
    v2f a;
    a.x = xr[kb + 2 * hf + 0];
    a.y = xr[kb + 2 * hf + 1];
#pragma unroll
    for (int nt = 0; nt < 4; ++nt) {
      // B fragment (4x16 f32): VGPR0=row(2*hf), VGPR1=row(2*hf+1), col = nt*16+lr
      v2f b;
      b.x = W[(kb + 2 * hf + 0) * DD + nt * 16 + lr];
      b.y = W[(kb + 2 * hf + 1) * DD + nt * 16 + lr];
      acc[nt] = __builtin_amdgcn_wmma_f32_16x16x4_f32(
          /*neg_a=*/false, a, /*neg_b=*/false, b,
          /*c_mod=*/(short)0, acc[nt], /*reuse_a=*/false, /*reuse_b=*/false);
    }
  }

#pragma unroll
  for (int nt = 0; nt < 4; ++nt) {
#pragma unroll
    for (int r = 0; r < 8; ++r) {
      int row = m0 + r + 8 * hf;  // C/D: VGPR r holds rows r (lanes 0-15), r+8 (16-31)
      if (row < N) {
        size_t idx = (size_t)row * DD + nt * 16 + lr;
        float v = acc[nt][r];
        if (Add) v += Add[idx];
        Y[idx] = v;
      }
    }
  }
}

// ------------------------- P1: edge logits + segment max ----------------------
__global__ __launch_bounds__(256) void k_edge_logits(
    const float* __restrict__ uh, const float* __restrict__ ih,
    const float* __restrict__ u_pos_k, const float* __restrict__ i_pos_k,
    const float* __restrict__ freq, const float* __restrict__ phase,
    const float* __restrict__ e_time, const int* __restrict__ e_iid,
    const int* __restrict__ e_uid, const int* __restrict__ pos_u,
    const int* __restrict__ pos_i, float* __restrict__ lg,
    unsigned* __restrict__ umaxL, unsigned* __restrict__ umaxS,
    unsigned* __restrict__ imaxL, unsigned* __restrict__ imaxS, int E) {
  int e = (blockIdx.x * 256 + threadIdx.x) >> 5;
  int lane = threadIdx.x & 31;
  if (e >= E) return;
  int iid = e_iid[e], uid = e_uid[e], pu = pos_u[e], pi = pos_i[e];
  float t = e_time[e];
  float sUL = 0.f, sIL = 0.f, sS = 0.f;
#pragma unroll
  for (int j = 0; j < 2; ++j) {
    int d = lane * 2 + j;
    float hi = ih[(size_t)iid * DD + d];
    float hu = uh[(size_t)uid * DD + d];
    float te = cosf(t * freq[d] + phase[d]);
    sS += hi * hu;
    sUL += (u_pos_k[pu * DD + d] + te) * hu;  // hi*hu part folded via sS
    sIL += (i_pos_k[pi * DD + d] + te) * hi;
  }
  sUL = wave_sum32(sUL);
  sIL = wave_sum32(sIL);
  sS = wave_sum32(sS);
  if (lane == 0) {
    const float inv = 0.125f;  // 1/sqrt(64)
    float uL = (sS + sUL) * inv;
    float iL = (sS + sIL) * inv;
    float S = sS * inv;
    size_t b = (size_t)e * 4;
    lg[b + 0] = uL;
    lg[b + 1] = iL;
    lg[b + 2] = S;
    atomicMax(&umaxL[uid], fkey(uL));
    atomicMax(&umaxS[uid], fkey(S));
    atomicMax(&imaxL[iid], fkey(iL));
    atomicMax(&imaxS[iid], fkey(S));
  }
}

// ------------------------- P2: exp + segment denominators ---------------------
__global__ __launch_bounds__(256) void k_edge_exp(
    const int* __restrict__ e_iid, const int* __restrict__ e_uid,
    float* __restrict__ lg, const unsigned* __restrict__ umaxL,
    const unsigned* __restrict__ umaxS, const unsigned* __restrict__ imaxL,
    const unsigned* __restrict__ imaxS, float* __restrict__ udenL,
    float* __restrict__ udenS, float* __restrict__ idenL,
    float* __restrict__ idenS, int E) {
  int e = blockIdx.x * 256 + threadIdx.x;
  if (e >= E) return;
  int iid = e_iid[e], uid = e_uid[e];
  size_t b = (size_t)e * 4;
  float uL = lg[b + 0], iL = lg[b + 1], S = lg[b + 2];
  float euL = expf(uL - funkey(umaxL[uid]));
  float eiL = expf(iL - funkey(imaxL[iid]));
  float euS = expf(S - funkey(umaxS[uid]));
  float eiS = expf(S - funkey(imaxS[iid]));
  lg[b + 0] = euL;
  lg[b + 1] = eiL;
  lg[b + 2] = euS;
  lg[b + 3] = eiS;
  atomicAdd(&udenL[uid], euL);
  atomicAdd(&udenS[uid], euS);
  atomicAdd(&idenL[iid], eiL);
  atomicAdd(&idenS[iid], eiS);
}

// ------------------------- P3: normalized weighted scatter --------------------
__global__ __launch_bounds__(256) void k_edge_scatter(
    const int* __restrict__ e_iid, const int* __restrict__ e_uid,
    const int* __restrict__ pos_u, const int* __restrict__ pos_i,
    const float* __restrict__ lg, const float* __restrict__ udenL,
    const float* __restrict__ udenS, const float* __restrict__ idenL,
    const float* __restrict__ idenS, const float* __restrict__ uh,
    const float* __restrict__ ih, const float* __restrict__ u_pos,
    const float* __restrict__ i_pos, float* __restrict__ hLu,
    float* __restrict__ hSu, float* __restrict__ hLi, float* __restrict__ hSi,
    int E) {
  long idx = (long)blockIdx.x * 256 + threadIdx.x;
  int e = (int)(idx >> 6);
  int d = (int)(idx & 63);
  if (e >= E) return;
  int iid = e_iid[e], uid = e_uid[e], pu = pos_u[e], pi = pos_i[e];
  size_t b = (size_t)e * 4;
  float aLu = lg[b + 0] / (udenL[uid] + 1e-9f);
  float aLi = lg[b + 1] / (idenL[iid] + 1e-9f);
  float aSu = lg[b + 2] / (udenS[uid] + 1e-9f);
  float aSi = lg[b + 3] / (idenS[iid] + 1e-9f);
  float hi = ih[(size_t)iid * DD + d];
  float hu = uh[(size_t)uid * DD + d];
  atomicAdd(&hLu[(size_t)uid * DD + d], aLu * (hi + u_pos[pu * DD + d]));
  atomicAdd(&hSu[(size_t)uid * DD + d], aSu * hi);
  atomicAdd(&hLi[(size_t)iid * DD + d], aLi * (hu + i_pos[pi * DD + d]));
  atomicAdd(&hSi[(size_t)iid * DD + d], aSi * hu);
}

// ------------------------- relation mean aggregation --------------------------
__global__ __launch_bounds__(256) void k_rel_scatter(const float* __restrict__ feat,
                                                     const int* __restrict__ src,
                                                     const int* __restrict__ dst,
                                                     float* __restrict__ sacc,
                                                     float* __restrict__ cnt, int E) {
  long idx = (long)blockIdx.x * 256 + threadIdx.x;
  int e = (int)(idx >> 6);
  int d = (int)(idx & 63);
  if (e >= E) return;
  int s = src[e], t = dst[e];
  atomicAdd(&sacc[(size_t)t * DD + d], feat[(size_t)s * DD + d]);
  if (d == 0) atomicAdd(&cnt[t], 1.0f);
}

__global__ __launch_bounds__(256) void k_rel_mean(float* __restrict__ sacc,
                                                  const float* __restrict__ cnt,
                                                  int Nrows) {
  long idx = (long)blockIdx.x * 256 + threadIdx.x;
  int i = (int)(idx >> 6);
  if (i >= Nrows) return;
  sacc[idx] = sacc[idx] / fmaxf(cnt[i], 1.0f);
}

// ------------------------- ELU residual ---------------------------------------
__global__ __launch_bounds__(256) void k_elu_res(float* __restrict__ out,
                                                 const float* __restrict__ old,
                                                 int n) {
  int i = blockIdx.x * 256 + threadIdx.x;
  if (i >= n) return;
  float x = out[i] + old[i];
  out[i] = x > 0.f ? x : expm1f(x);
}

// ------------------------- host launcher --------------------------------------
extern "C" void kernel_launch(void* const* d_in, const int* in_sizes, int n_in,
                              void* d_out, int out_size, void* d_ws, size_t ws_size,
                              hipStream_t stream) {
  (void)in_sizes; (void)n_in; (void)out_size; (void)ws_size;
  const float* user_h = (const float*)d_in[0];
  const float* item_h = (const float*)d_in[1];
  const float* brand_feat = (const float*)d_in[2];
  const float* cat_feat = (const float*)d_in[3];
  const float* rel_feat = (const float*)d_in[4];
  const float* Wu = (const float*)d_in[5];
  const float* Wi = (const float*)d_in[6];
  const float* Wb = (const float*)d_in[7];
  const float* Wc = (const float*)d_in[8];
  const float* Wr = (const float*)d_in[9];
  const float* Wg_u = (const float*)d_in[10];
  const float* Wg_i = (const float*)d_in[11];
  const float* u_pos = (const float*)d_in[12];
  const float* u_pos_k = (const float*)d_in[13];
  const float* i_pos = (const float*)d_in[14];
  const float* i_pos_k = (const float*)d_in[15];
  const float* basis_freq = (const float*)d_in[16];
  const float* phase = (const float*)d_in[17];
  const float* Wr0 = (const float*)d_in[18];
  const float* Wr1 = (const float*)d_in[19];
  const float* Wr2 = (const float*)d_in[20];
  const float* e_time = (const float*)d_in[21];
  const int* e_iid = (const int*)d_in[22];
  const int* e_uid = (const int*)d_in[23];
  const int* pos_u = (const int*)d_in[24];
  const int* pos_i = (const int*)d_in[25];
  const int* eb_src = (const int*)d_in[26];
  const int* eb_dst = (const int*)d_in[27];
  const int* ec_src = (const int*)d_in[28];
  const int* ec_dst = (const int*)d_in[29];
  const int* er_src = (const int*)d_in[30];
  const int* er_dst = (const int*)d_in[31];

  // bump allocator over d_ws
  char* wsb = (char*)d_ws;
  size_t off = 0;
  auto alloc = [&](size_t bytes) -> void* {
    void* p = wsb + off;
    off = (off + bytes + 255) & ~(size_t)255;
    return p;
  };
  float* uh = (float*)alloc((size_t)cNu * DD * 4);
  float* ih = (float*)alloc((size_t)cNi * DD * 4);
  float* bf = (float*)alloc((size_t)cNb * DD * 4);
  float* cf = (float*)alloc((size_t)cNc * DD * 4);
  float* rf = (float*)alloc((size_t)cNr * DD * 4);
  float* lg = (float*)alloc((size_t)cE * 4 * 4);
  unsigned* umaxL = (unsigned*)alloc((size_t)cNu * 4);
  unsigned* umaxS = (unsigned*)alloc((size_t)cNu * 4);
  unsigned* imaxL = (unsigned*)alloc((size_t)cNi * 4);
  unsigned* imaxS = (unsigned*)alloc((size_t)cNi * 4);
  float* udenL = (float*)alloc((size_t)cNu * 4);
  float* udenS = (float*)alloc((size_t)cNu * 4);
  float* idenL = (float*)alloc((size_t)cNi * 4);
  float* idenS = (float*)alloc((size_t)cNi * 4);
  float* hLu = (float*)alloc((size_t)cNu * DD * 4);
  float* hSu = (float*)alloc((size_t)cNu * DD * 4);
  float* hLi = (float*)alloc((size_t)cNi * DD * 4);
  float* hSi = (float*)alloc((size_t)cNi * DD * 4);
  float* sacc = (float*)alloc((size_t)cNi * DD * 4);
  float* cnt = (float*)alloc((size_t)cNi * 4);

  auto nb = [](long n) { return (int)((n + 255) / 256); };
  auto gb = [](int N) { return (N + 127) / 128; };  // gemm: 128 rows / block

  // ---- init ----
  k_fill_u32<<<nb(cNu), 256, 0, stream>>>(umaxL, cNu, NEGINF_KEY);
  k_fill_u32<<<nb(cNu), 256, 0, stream>>>(umaxS, cNu, NEGINF_KEY);
  k_fill_u32<<<nb(cNi), 256, 0, stream>>>(imaxL, cNi, NEGINF_KEY);
  k_fill_u32<<<nb(cNi), 256, 0, stream>>>(imaxS, cNi, NEGINF_KEY);
  k_fill_f32<<<nb(cNu), 256, 0, stream>>>(udenL, cNu, 0.f);
  k_fill_f32<<<nb(cNu), 256, 0, stream>>>(udenS, cNu, 0.f);
  k_fill_f32<<<nb(cNi), 256, 0, stream>>>(idenL, cNi, 0.f);
  k_fill_f32<<<nb(cNi), 256, 0, stream>>>(idenS, cNi, 0.f);
  k_fill_f32<<<nb((long)cNu * DD), 256, 0, stream>>>(hLu, cNu * DD, 0.f);
  k_fill_f32<<<nb((long)cNu * DD), 256, 0, stream>>>(hSu, cNu * DD, 0.f);
  k_fill_f32<<<nb((long)cNi * DD), 256, 0, stream>>>(hLi, cNi * DD, 0.f);
  k_fill_f32<<<nb((long)cNi * DD), 256, 0, stream>>>(hSi, cNi * DD, 0.f);

  // ---- relation-specific dense transforms (WMMA GEMMs) ----
  k_gemm64<<<gb(cNu), 256, 0, stream>>>(user_h, Wu, nullptr, uh, cNu);
  k_gemm64<<<gb(cNi), 256, 0, stream>>>(item_h, Wi, nullptr, ih, cNi);
  k_gemm64<<<gb(cNb), 256, 0, stream>>>(brand_feat, Wb, nullptr, bf, cNb);
  k_gemm64<<<gb(cNc), 256, 0, stream>>>(cat_feat, Wc, nullptr, cf, cNc);
  k_gemm64<<<gb(cNr), 256, 0, stream>>>(rel_feat, Wr, nullptr, rf, cNr);

  // ---- edge attention: logits/max -> exp/denom -> normalized scatter ----
  k_edge_logits<<<nb((long)cE * 32), 256, 0, stream>>>(
      uh, ih, u_pos_k, i_pos_k, basis_freq, phase, e_time, e_iid, e_uid, pos_u,
      pos_i, lg, umaxL, umaxS, imaxL, imaxS, cE);
  k_edge_exp<<<nb(cE), 256, 0, stream>>>(e_iid, e_uid, lg, umaxL, umaxS, imaxL,
                                         imaxS, udenL, udenS, idenL, idenS, cE);
  k_edge_scatter<<<nb((long)cE * 64), 256, 0, stream>>>(
      e_iid, e_uid, pos_u, pos_i, lg, udenL, udenS, idenL, idenS, uh, ih, u_pos,
      i_pos, hLu, hSu, hLi, hSi, cE);

  // ---- gate GEMMs: concat([hL,hS]) @ Wg == hL@Wg[0:64] + hS@Wg[64:128] ----
  float* out_u = (float*)d_out;
  float* out_i = out_u + (size_t)cNu * DD;
  k_gemm64<<<gb(cNu), 256, 0, stream>>>(hLu, Wg_u, nullptr, out_u, cNu);
  k_gemm64<<<gb(cNu), 256, 0, stream>>>(hSu, Wg_u + 64 * DD, out_u, out_u, cNu);
  k_gemm64<<<gb(cNi), 256, 0, stream>>>(hLi, Wg_i, nullptr, out_i, cNi);
  k_gemm64<<<gb(cNi), 256, 0, stream>>>(hSi, Wg_i + 64 * DD, out_i, out_i, cNi);

  // ---- relation mean aggregations, accumulated into item output ----
  const float* rfeat[3] = {bf, cf, rf};
  const int* rsrc[3] = {eb_src, ec_src, er_src};
  const int* rdst[3] = {eb_dst, ec_dst, er_dst};
  const float* rW[3] = {Wr0, Wr1, Wr2};
  const int rne[3] = {cEb, cEc, cEr};
  for (int r = 0; r < 3; ++r) {
    k_fill_f32<<<nb((long)cNi * DD), 256, 0, stream>>>(sacc, cNi * DD, 0.f);
    k_fill_f32<<<nb(cNi), 256, 0, stream>>>(cnt, cNi, 0.f);
    k_rel_scatter<<<nb((long)rne[r] * 64), 256, 0, stream>>>(rfeat[r], rsrc[r],
                                                             rdst[r], sacc, cnt,
                                                             rne[r]);
    k_rel_mean<<<nb((long)cNi * DD), 256, 0, stream>>>(sacc, cnt, cNi);
    k_gemm64<<<gb(cNi), 256, 0, stream>>>(sacc, rW[r], out_i, out_i, cNi);
  }

  // ---- residual ELU ----
  k_elu_res<<<nb((long)cNu * DD), 256, 0, stream>>>(out_u, user_h, cNu * DD);
  k_elu_res<<<nb((long)cNi * DD), 256, 0, stream>>>(out_i, item_h, cNi * DD);
}